// OneLayerLSTM_1305670058710
// MI455X (gfx1250) — compile-verified
//
#include <hip/hip_runtime.h>

// ---------------------------------------------------------------------------
// Bidirectional LSTM (B=128, T=4096, F=80, H=64) + FC(2H->8) for MI455X.
// Strategy: 16 workgroups = 8 batch-tiles x 2 directions; each WG runs the
// whole T-step recurrence for its 16 batch rows using V_WMMA_F32_16X16X4_F32.
// Weight (B-matrix) fragments are preloaded into VGPRs; per-step A-fragments
// come from LDS (x tile + previous h). Cell state c lives in registers.
// ---------------------------------------------------------------------------

typedef __attribute__((ext_vector_type(2))) float v2f;
typedef __attribute__((ext_vector_type(4))) float v4f;
typedef __attribute__((ext_vector_type(8))) float v8f;

#define BB 128
#define TT 4096
#define FF 80
#define HH 64

// padded LDS strides (floats); all are even and row-stride*4 is 16B-multiple
#define XS 84    // 84*4 = 336 = 21*16
#define HS 68    // 68*4 = 272 = 17*16
#define GS 260   // 260*4 = 1040 = 65*16

__device__ __forceinline__ float sigm(float v) {
    return 1.0f / (1.0f + __expf(-v));
}

__global__ __launch_bounds__(256, 1)
void lstm_dir_kernel(const float* __restrict__ x,
                     const float* __restrict__ w_ih_f, const float* __restrict__ w_hh_f,
                     const float* __restrict__ b_ih_f, const float* __restrict__ b_hh_f,
                     const float* __restrict__ w_ih_b, const float* __restrict__ w_hh_b,
                     const float* __restrict__ b_ih_b, const float* __restrict__ b_hh_b,
                     float* __restrict__ hbuf)   // [2][B][H] in workspace
{
    const int tile = blockIdx.x;          // batch tile: rows [16*tile, 16*tile+16)
    const int dir  = blockIdx.y;          // 0 = forward, 1 = backward (flipped time)
    const int tid  = threadIdx.x;         // 256 threads = 8 wave32
    const int wave = tid >> 5;            // 0..7 -> gate columns [32w, 32w+32)
    const int lane = tid & 31;
    const int m16  = lane & 15;           // row within 16x16 tile (A/M index)
    const int hi   = lane >> 4;           // lane half selects K pair (ISA A layout)

    const float* w_ih = dir ? w_ih_b : w_ih_f;   // [256, 80]
    const float* w_hh = dir ? w_hh_b : w_hh_f;   // [256, 64]
    const float* b_ih = dir ? b_ih_b : b_ih_f;   // [256]
    const float* b_hh = dir ? b_hh_b : b_hh_f;   // [256]

    __shared__ float xs[16 * XS];   // x_t tile [16 x 80]
    __shared__ float hs[16 * HS];   // h_{t-1} tile [16 x 64]
    __shared__ float gs[16 * GS];   // activated gates [16 x 256]

    // Gate columns owned by this wave's two 16-wide N tiles.
    const int n0 = wave * 32 + m16;
    const int n1 = n0 + 16;

    // ---- Preload B-matrix fragments (w.T columns) into VGPRs -------------
    // B layout assumption (mirror of documented A layout): fragment .x/.y hold
    // rows K = k0+2*hi, k0+1+2*hi for column n = lane%16 of the N-tile.
    v2f bih0[20], bih1[20];   // K = 80 -> 20 K-steps
    v2f bhh0[16], bhh1[16];   // K = 64 -> 16 K-steps
#pragma unroll
    for (int kk = 0; kk < 20; ++kk) {
        const int k = 4 * kk + 2 * hi;
        bih0[kk] = *(const v2f*)(w_ih + n0 * FF + k);
        bih1[kk] = *(const v2f*)(w_ih + n1 * FF + k);
    }
#pragma unroll
    for (int kk = 0; kk < 16; ++kk) {
        const int k = 4 * kk + 2 * hi;
        bhh0[kk] = *(const v2f*)(w_hh + n0 * HH + k);
        bhh1[kk] = *(const v2f*)(w_hh + n1 * HH + k);
    }
    const float bias0 = b_ih[n0] + b_hh[n0];
    const float bias1 = b_ih[n1] + b_hh[n1];

    // zero initial hidden state
    for (int i = tid; i < 16 * HS; i += 256) hs[i] = 0.0f;

    // Combiner role: each thread owns 4 contiguous (row, k) cells of [16x64];
    // cell state c stays in registers for all T steps.
    const int cell = tid * 4;
    const int cm = cell >> 6;         // batch row within tile
    const int ck = cell & 63;         // hidden index (multiple of 4)
    float c0 = 0.f, c1 = 0.f, c2 = 0.f, c3 = 0.f;

    // waves 0-3: i,f gates (sigmoid); 4-5: g gate (tanh); 6-7: o gate (sigmoid)
    const bool use_tanh = ((wave >> 1) == 2);

    const float* xrow_base = x + (size_t)(tile * 16) * TT * FF;
    const int abase_x = m16 * XS + 2 * hi;
    const int abase_h = m16 * HS + 2 * hi;

#pragma unroll 1
    for (int step = 0; step < TT; ++step) {
        const int t = dir ? (TT - 1 - step) : step;
        __syncthreads();   // h_{t-1} in LDS; previous x/gates consumed

        // ---- cooperative load x_t tile (16x80 fp32) into LDS ------------
#pragma unroll
        for (int i = 0; i < 5; ++i) {
            const int idx = tid + i * 256;          // 256*5 == 1280 == 16*80
            const int r = idx / FF, cc = idx - r * FF;
            xs[r * XS + cc] = xrow_base[(size_t)r * TT * FF + (size_t)t * FF + cc];
        }
        // prefetch next timestep's rows (global_prefetch_b8); 128B lines at
        // offsets 0/128/256 of each 320B row stay inside the allocation.
        if (step + 1 < TT) {
            const int tn = dir ? (TT - 2 - step) : (step + 1);
            const int chunk = tid >> 4;             // 0..15
            if (chunk < 3) {
                const char* p = (const char*)(xrow_base + (size_t)(tid & 15) * TT * FF
                                              + (size_t)tn * FF) + chunk * 128;
                __builtin_prefetch(p, 0, 1);
            }
        }
        __syncthreads();   // x_t ready

        // ---- gates = bias + x_t @ w_ih.T + h @ w_hh.T  (fp32 WMMA) ------
        v8f acc0, acc1;
#pragma unroll
        for (int j = 0; j < 8; ++j) { acc0[j] = bias0; acc1[j] = bias1; }

#pragma unroll
        for (int kk = 0; kk < 20; ++kk) {
            const v2f a = *(const v2f*)(&xs[abase_x + 4 * kk]);
            acc0 = __builtin_amdgcn_wmma_f32_16x16x4_f32(false, a, false, bih0[kk],
                                                         (short)0, acc0, false, false);
            acc1 = __builtin_amdgcn_wmma_f32_16x16x4_f32(false, a, false, bih1[kk],
                                                         (short)0, acc1, false, false);
        }
#pragma unroll
        for (int kk = 0; kk < 16; ++kk) {
            const v2f a = *(const v2f*)(&hs[abase_h + 4 * kk]);
            acc0 = __builtin_amdgcn_wmma_f32_16x16x4_f32(false, a, false, bhh0[kk],
                                                         (short)0, acc0, false, false);
            acc1 = __builtin_amdgcn_wmma_f32_16x16x4_f32(false, a, false, bhh1[kk],
                                                         (short)0, acc1, false, false);
        }

        // ---- wave-uniform activation, scatter to gates LDS --------------
        // C/D layout: VGPR j holds row M = j + 8*hi, column n = lane%16.
#pragma unroll
        for (int j = 0; j < 8; ++j) {
            const int M = j + 8 * hi;
            float g0 = acc0[j], g1 = acc1[j];
            if (use_tanh) { g0 = tanhf(g0); g1 = tanhf(g1); }
            else          { g0 = sigm(g0);  g1 = sigm(g1);  }
            gs[M * GS + n0] = g0;
            gs[M * GS + n1] = g1;
        }
        __syncthreads();   // activated gates ready

        // ---- combiner: c = f*c + i*g ; h = o*tanh(c) --------------------
        const v4f iv = *(const v4f*)(&gs[cm * GS +        ck]);
        const v4f fv = *(const v4f*)(&gs[cm * GS +  64 + ck]);
        const v4f gv = *(const v4f*)(&gs[cm * GS + 128 + ck]);
        const v4f ov = *(const v4f*)(&gs[cm * GS + 192 + ck]);
        c0 = fv[0] * c0 + iv[0] * gv[0];
        c1 = fv[1] * c1 + iv[1] * gv[1];
        c2 = fv[2] * c2 + iv[2] * gv[2];
        c3 = fv[3] * c3 + iv[3] * gv[3];
        v4f hv;
        hv[0] = ov[0] * tanhf(c0);
        hv[1] = ov[1] * tanhf(c1);
        hv[2] = ov[2] * tanhf(c2);
        hv[3] = ov[3] * tanhf(c3);
        *(v4f*)(&hs[cm * HS + ck]) = hv;
    }

    __syncthreads();
    // final hidden state -> workspace: hbuf[dir][b][k]
    const v4f hv = *(const v4f*)(&hs[cm * HS + ck]);
    *(v4f*)(&hbuf[(size_t)dir * BB * HH + (size_t)(tile * 16 + cm) * HH + ck]) = hv;
}

// out[b,o] = b_fc[o] + h_fwd[b,:] . w_fc[o,0:64] + h_bwd[b,:] . w_fc[o,64:128]
__global__ void fc_kernel(const float* __restrict__ hbuf,
                          const float* __restrict__ w_fc,   // [8, 128]
                          const float* __restrict__ b_fc,   // [8]
                          float* __restrict__ out)          // [128, 8]
{
    const int idx = blockIdx.x * 256 + threadIdx.x;
    if (idx >= BB * 8) return;
    const int b = idx >> 3, o = idx & 7;
    const float* hf = hbuf + (size_t)b * HH;
    const float* hb = hbuf + (size_t)BB * HH + (size_t)b * HH;
    const float* wf = w_fc + o * 128;
    float s = b_fc[o];
#pragma unroll 8
    for (int k = 0; k < HH; ++k)
        s = fmaf(hf[k], wf[k], fmaf(hb[k], wf[64 + k], s));
    out[idx] = s;
}

extern "C" void kernel_launch(void* const* d_in, const int* in_sizes, int n_in,
                              void* d_out, int out_size, void* d_ws, size_t ws_size,
                              hipStream_t stream) {
    const float* x      = (const float*)d_in[0];
    // d_in[1] = lengths (unused by the reference forward pass)
    const float* w_ih_f = (const float*)d_in[2];
    const float* w_hh_f = (const float*)d_in[3];
    const float* b_ih_f = (const float*)d_in[4];
    const float* b_hh_f = (const float*)d_in[5];
    const float* w_ih_b = (const float*)d_in[6];
    const float* w_hh_b = (const float*)d_in[7];
    const float* b_ih_b = (const float*)d_in[8];
    const float* b_hh_b = (const float*)d_in[9];
    const float* w_fc   = (const float*)d_in[10];
    const float* b_fc   = (const float*)d_in[11];
    float* out  = (float*)d_out;
    float* hbuf = (float*)d_ws;                    // 2*128*64 fp32 = 64 KB

    dim3 grid1(BB / 16, 2);                        // 8 batch tiles x 2 directions
    lstm_dir_kernel<<<grid1, 256, 0, stream>>>(x, w_ih_f, w_hh_f, b_ih_f, b_hh_f,
                                               w_ih_b, w_hh_b, b_ih_b, b_hh_b, hbuf);
    fc_kernel<<<(BB * 8 + 255) / 256, 256, 0, stream>>>(hbuf, w_fc, b_fc, out);
}